// GraphDenseGinNet_36850819400349
// MI455X (gfx1250) — compile-verified
//
#include <hip/hip_runtime.h>

#define N_NODES  50000
#define N_EDGES  400000
#define N_GRAPHS 256
#define OUT_DIM  128
#define F_IN     78
#define BN_EPS   1e-5f
#define MT       4          // M-tiles (16 rows each) per wave

typedef float v2f __attribute__((ext_vector_type(2)));
typedef float v8f __attribute__((ext_vector_type(8)));

static inline int cdiv(int a, int b) { return (a + b - 1) / b; }

// ---------------------------------------------------------------------------
// D[r, 0:C] = S[r, 0:C]   (strided slice copy)
// ---------------------------------------------------------------------------
__global__ void k_copy_slice(const float* __restrict__ S, int lds,
                             float* __restrict__ D, int ldd, int M, int C)
{
    int id = blockIdx.x * blockDim.x + threadIdx.x;
    if (id >= M * C) return;
    int r = id / C, c = id - r * C;
    D[r * ldd + c] = S[r * lds + c];
}

// ---------------------------------------------------------------------------
// T[dst[e], c] += X[src[e], c]   (edge scatter-add; features are L2-resident)
// ---------------------------------------------------------------------------
__global__ void k_edge_scatter(const float* __restrict__ X, int ldx,
                               float* __restrict__ T, int ldt,
                               const int* __restrict__ src,
                               const int* __restrict__ dst,
                               int nE, int C)
{
    int id = blockIdx.x * blockDim.x + threadIdx.x;
    if (id >= nE * C) return;
    int e = id / C, c = id - e * C;
    atomicAdd(&T[dst[e] * ldt + c], X[src[e] * ldx + c]);
}

// ---------------------------------------------------------------------------
// WMMA f32 GEMM:  Y[M,Cout] = act(X[M,K] @ W[K,Cout] + bias)
// One wave32 computes a 64x16 output strip (MT=4 stacked 16x16 tiles) with
// V_WMMA_F32_16X16X4_F32, sharing one B fragment across the 4 tiles.
//   A frag (16x4 f32, 2 VGPRs): lane L -> M=L&15, K = 2*(L>>4) + v
//   B frag (4x16 f32, 2 VGPRs): lane L -> N=L&15, K = 2*(L>>4) + v
//   C/D   (16x16 f32, 8 VGPRs): lane L, vgpr v -> M = 8*(L>>4)+v, N = L&15
// Requirements: M % 16 == 0 (50000, 256 ok); ldx even (all dims even here).
// Main K loop is guard-free; single clamped tail handles K % 4 != 0.
// Out-of-range columns / clamped rows read valid finite data whose products
// land only in never-stored outputs (or are multiplied by zero A in the tail).
// ---------------------------------------------------------------------------
__global__ __launch_bounds__(128)
void k_gemm_wmma(const float* __restrict__ X, int ldx,
                 const float* __restrict__ W,
                 const float* __restrict__ bias,
                 float* __restrict__ Y, int ldy,
                 int M, int K, int Cout, int relu)
{
    const int lane   = threadIdx.x & 31;
    const int wv     = threadIdx.x >> 5;
    const int tilesN = (Cout + 15) >> 4;
    const int tilesM = M >> 4;
    const int nGrp   = (tilesM + MT - 1) / MT;
    const int tiles  = nGrp * tilesN;
    const int tile   = blockIdx.x * 4 + wv;          // wave-uniform
    if (tile >= tiles) return;

    const int tg   = tile / tilesN;                  // M-tile group
    const int tn   = tile - tg * tilesN;             // N-tile (fastest: waves share A)
    const int col  = tn * 16 + (lane & 15);
    const bool cok = (col < Cout);
    const int colc = cok ? col : (Cout - 1);
    const int kofs = (lane >> 4) << 1;

    // Per-subtile row pointers; clamp overflow rows to a valid row.
    const float* Xr[MT];
    int rb[MT];
#pragma unroll
    for (int s = 0; s < MT; ++s) {
        rb[s] = (tg * MT + s) * 16;
        int r = rb[s] + (lane & 15);
        if (r > M - 1) r = M - 1;
        Xr[s] = X + r * ldx;
    }

    v8f acc[MT];
#pragma unroll
    for (int s = 0; s < MT; ++s)
        acc[s] = v8f{0.f, 0.f, 0.f, 0.f, 0.f, 0.f, 0.f, 0.f};

    const float* Wp = W + colc;
    int kk = 0;
    for (; kk + 4 <= K; kk += 4) {                   // guard-free main loop
        const int k0 = kk + kofs;
        v2f b;
        b.x = Wp[k0 * Cout];
        b.y = Wp[(k0 + 1) * Cout];
#pragma unroll
        for (int s = 0; s < MT; ++s) {
            v2f a = *(const v2f*)(Xr[s] + k0);       // single b64 load
            acc[s] = __builtin_amdgcn_wmma_f32_16x16x4_f32(
                false, a, false, b, (short)0, acc[s], false, false);
        }
    }
    if (kk < K) {                                    // tail: K % 4 != 0 (K = 78)
        const int k0  = kk + kofs;
        const int kc0 = (k0     < K) ? k0     : (K - 1);
        const int kc1 = (k0 + 1 < K) ? k0 + 1 : (K - 1);
        v2f b;
        b.x = Wp[kc0 * Cout];
        b.y = Wp[kc1 * Cout];
#pragma unroll
        for (int s = 0; s < MT; ++s) {
            v2f a;
            a.x = (k0     < K) ? Xr[s][k0]     : 0.f;
            a.y = (k0 + 1 < K) ? Xr[s][k0 + 1] : 0.f;
            acc[s] = __builtin_amdgcn_wmma_f32_16x16x4_f32(
                false, a, false, b, (short)0, acc[s], false, false);
        }
    }

    if (!cok) return;
    const float bv = bias[col];
    const int rofs = (lane >> 4) << 3;
#pragma unroll
    for (int s = 0; s < MT; ++s) {
        if (rb[s] >= M) break;                       // M % 16 == 0: whole-tile guard
        const int r0 = rb[s] + rofs;
#pragma unroll
        for (int v = 0; v < 8; ++v) {
            float o = acc[s][v] + bv;
            if (relu) o = fmaxf(o, 0.f);
            Y[(r0 + v) * ldy + col] = o;
        }
    }
}

// ---------------------------------------------------------------------------
// Per-channel sum / sum-of-squares over M rows (H contiguous [M,C], C<=128)
// ---------------------------------------------------------------------------
__global__ void k_bn_stats(const float* __restrict__ H, int C, int M,
                           float* __restrict__ stats /* [2*C] */)
{
    int c = threadIdx.x;
    if (c >= C) return;
    int r0 = blockIdx.x * 128;
    int r1 = r0 + 128; if (r1 > M) r1 = M;
    float s = 0.f, s2 = 0.f;
    for (int r = r0; r < r1; ++r) {
        float v = H[r * C + c];
        s += v; s2 += v * v;
    }
    atomicAdd(&stats[c], s);
    atomicAdd(&stats[C + c], s2);
}

// ---------------------------------------------------------------------------
// Y[r, yoff+c] = relu((H[r,c]-mu)*rsqrt(var+eps)*gamma + beta)
// ---------------------------------------------------------------------------
__global__ void k_bn_apply(const float* __restrict__ H,
                           const float* __restrict__ stats,
                           const float* __restrict__ gamma,
                           const float* __restrict__ beta,
                           float* __restrict__ Y, int ldy, int yoff,
                           int M, int C)
{
    int id = blockIdx.x * blockDim.x + threadIdx.x;
    if (id >= M * C) return;
    int r = id / C, c = id - r * C;
    const float invM = 1.0f / (float)M;
    float mu  = stats[c] * invM;
    float var = stats[C + c] * invM - mu * mu;
    float rs  = rsqrtf(var + BN_EPS);
    float o = (H[id] - mu) * rs * gamma[c] + beta[c];
    Y[r * ldy + yoff + c] = fmaxf(o, 0.f);
}

// ---------------------------------------------------------------------------
// Mean pooling over graphs
// ---------------------------------------------------------------------------
__global__ void k_count(const int* __restrict__ batch, float* __restrict__ cnt, int M)
{
    int id = blockIdx.x * blockDim.x + threadIdx.x;
    if (id < M) atomicAdd(&cnt[batch[id]], 1.0f);
}

__global__ void k_pool(const float* __restrict__ H, int ld,
                       const int* __restrict__ batch,
                       float* __restrict__ pool, int M, int C)
{
    int id = blockIdx.x * blockDim.x + threadIdx.x;
    if (id >= M * C) return;
    int r = id / C, c = id - r * C;
    atomicAdd(&pool[batch[r] * C + c], H[r * ld + c]);
}

__global__ void k_pool_fin(const float* __restrict__ pool,
                           const float* __restrict__ cnt,
                           float* __restrict__ out, int G, int C)
{
    int id = blockIdx.x * blockDim.x + threadIdx.x;
    if (id >= G * C) return;
    int g = id / C;
    out[id] = pool[id] / fmaxf(cnt[g], 1.0f);
}

// ===========================================================================
extern "C" void kernel_launch(void* const* d_in, const int* in_sizes, int n_in,
                              void* d_out, int out_size, void* d_ws, size_t ws_size,
                              hipStream_t stream)
{
    (void)n_in; (void)out_size; (void)ws_size;

    const float* x     = (const float*)d_in[0];
    const int*   ei    = (const int*)d_in[1];
    const int*   src   = ei;
    const int*   dst   = ei + N_EDGES;
    const int*   batch = (const int*)d_in[2];

    // ---- param leaf ordering detection (insertion vs sorted-dict flatten) ----
    const int NLAYER = 29;
    bool sortedTop = (in_sizes[3] == 92 * OUT_DIM);     // Wc (11776) first => sorted
    int layerBase  = sortedTop ? 5 : 3;
    const float* Wc = (const float*)d_in[sortedTop ? 3 : layerBase + NLAYER * 6];
    const float* bc = (const float*)d_in[sortedTop ? 4 : layerBase + NLAYER * 6 + 1];
    // layer0: ci=78, co=32. sorted per-layer: [W1,W2,b1,b2,beta,gamma] -> slot1 = 1024
    //         insertion:               [W1,b1,W2,b2,gamma,beta]       -> slot1 = 32
    bool sortedLayer = (in_sizes[layerBase + 1] == 32 * 32);
    int lmap[6];    // logical {W1,b1,W2,b2,gamma,beta} -> physical slot
    if (sortedLayer) { lmap[0]=0; lmap[1]=2; lmap[2]=1; lmap[3]=3; lmap[4]=5; lmap[5]=4; }
    else             { lmap[0]=0; lmap[1]=1; lmap[2]=2; lmap[3]=3; lmap[4]=4; lmap[5]=5; }
    auto LP = [&](int li, int which) -> const float* {
        return (const float*)d_in[layerBase + 6 * li + lmap[which]];
    };

    // ---- workspace layout (floats) ----
    float* CB    = (float*)d_ws;                 // concat features, 184*N
    float* T     = CB + 184 * N_NODES;           // aggregate buffer, 184*N
    float* H2    = T;                            // GEMM2 out aliases T (T dead by then)
    float* M1    = T + 184 * N_NODES;            // MLP hidden, 128*N
    float* G     = M1 + 128 * N_NODES;           // inner-GIN output, 128*N
    float* stats = G + 128 * N_NODES;            // 256
    float* pool  = stats + 256;                  // 256*92
    float* cnt   = pool + N_GRAPHS * 92;         // 256
    float* pooln = cnt + N_GRAPHS;               // 256*92

    auto launch_gemm = [&](const float* X, int ldx, const float* W,
                           const float* bias, float* Y, int ldy,
                           int M, int K, int Cout, int relu)
    {
        int tiles = cdiv(M >> 4, MT) * cdiv(Cout, 16);
        k_gemm_wmma<<<cdiv(tiles, 4), 128, 0, stream>>>(X, ldx, W, bias, Y, ldy,
                                                        M, K, Cout, relu);
    };

    // ---- one full GINConv+BN+ReLU layer ----
    auto rungin = [&](const float* in, int ldin, int Ci, int li, int Co,
                      float* out, int ldo, int ooff)
    {
        const float* W1 = LP(li, 0), *b1 = LP(li, 1);
        const float* W2 = LP(li, 2), *b2 = LP(li, 3);
        const float* ga = LP(li, 4), *be = LP(li, 5);
        // t = x + scatter_add(x[src] -> dst)
        k_copy_slice<<<cdiv(N_NODES * Ci, 256), 256, 0, stream>>>(in, ldin, T, Ci, N_NODES, Ci);
        k_edge_scatter<<<cdiv(N_EDGES * Ci, 256), 256, 0, stream>>>(in, ldin, T, Ci,
                                                                    src, dst, N_EDGES, Ci);
        // m = relu(t @ W1 + b1);  h = m @ W2 + b2
        launch_gemm(T, Ci, W1, b1, M1, Co, N_NODES, Ci, Co, 1);
        launch_gemm(M1, Co, W2, b2, H2, Co, N_NODES, Co, Co, 0);
        // BN (training-mode, biased var) + ReLU, written into destination slice
        hipMemsetAsync(stats, 0, 2 * Co * sizeof(float), stream);
        k_bn_stats<<<cdiv(N_NODES, 128), 128, 0, stream>>>(H2, Co, N_NODES, stats);
        k_bn_apply<<<cdiv(N_NODES * Co, 256), 256, 0, stream>>>(H2, stats, ga, be,
                                                                out, ldo, ooff, N_NODES, Co);
    };

    // ---- schedule ----
    const int blkW[4] = {128, 160, 176, 184};    // block concat widths
    const int bs[4]   = {2, 3, 4, 4};            // bn_sizes -> conv1 width = 32*bs
    int li = 0;

    // conv0: x[N,78] -> CB[:, 0:32) with block1 stride
    rungin(x, F_IN, F_IN, li++, 32, CB, blkW[0], 0);

    int c = 32;
    for (int b = 0; b < 4; ++b) {
        const int Wb  = blkW[b];
        const int co1 = 32 * bs[b];
        int curW = c;
        for (int i = 0; i < 3; ++i) {
            // DenseLayer.conv1: concat[N,curW] -> G[N,co1]
            rungin(CB, Wb, curW, li++, co1, G, co1, 0);
            // DenseLayer.conv2: G[N,co1] -> CB[:, curW:curW+32)
            rungin(G, co1, co1, li++, 32, CB, Wb, curW);
            curW += 32;
        }
        // transition: concat[N,Wb] -> CB[:, 0:Wb/2) with NEXT block's stride
        const int c2 = Wb / 2;
        const int ldNext = (b < 3) ? blkW[b + 1] : c2;
        rungin(CB, Wb, Wb, li++, c2, CB, ldNext, 0);
        c = c2;
    }
    // final features: CB [N, 92], ld = 92

    // ---- global mean pool over 256 graphs ----
    hipMemsetAsync(pool, 0, N_GRAPHS * 92 * sizeof(float), stream);
    hipMemsetAsync(cnt, 0, N_GRAPHS * sizeof(float), stream);
    k_count<<<cdiv(N_NODES, 256), 256, 0, stream>>>(batch, cnt, N_NODES);
    k_pool<<<cdiv(N_NODES * 92, 256), 256, 0, stream>>>(CB, 92, batch, pool, N_NODES, 92);
    k_pool_fin<<<cdiv(N_GRAPHS * 92, 256), 256, 0, stream>>>(pool, cnt, pooln, N_GRAPHS, 92);

    // ---- classifier: pooled[256,92] @ Wc[92,128] + bc -> d_out[256,128] ----
    launch_gemm(pooln, 92, Wc, bc, (float*)d_out, OUT_DIM, N_GRAPHS, 92, OUT_DIM, 0);
}